// SimpleKVCacheTest_2430951490279
// MI455X (gfx1250) — compile-verified
//
#include <hip/hip_runtime.h>
#include <math.h>

// Problem constants (from reference)
#define B_      16
#define H_      32
#define D_      128
#define PAST_   4095
#define LTOT    4096          // PAST_ + 1 keys after append
#define NSPLIT  8
#define KEYS_PER_SPLIT (LTOT / NSPLIT)        // 512
#define TILES_PER_SPLIT (KEYS_PER_SPLIT / 16) // 32
#define ROWSTRIDE (H_ * D_)   // 4096 floats between consecutive keys in cache
#define SK_PITCH (D_ + 4)     // LDS row pitch (floats); +16B breaks bank conflicts
#define SK_BUFBYTES (16 * SK_PITCH * 4)       // one K-tile buffer, bytes

typedef __attribute__((ext_vector_type(2))) float v2f;
typedef __attribute__((ext_vector_type(4))) float v4f;
typedef __attribute__((ext_vector_type(8))) float v8f;

// ---- CDNA5 async copy: global -> LDS, tracked by ASYNCcnt -----------------
__device__ __forceinline__ void async_ld_b128(unsigned lds_addr,
                                              const float* gaddr) {
    // GV mode: vdst = LDS byte address, vaddr = 64-bit global address
    asm volatile("global_load_async_to_lds_b128 %0, %1, off"
                 :: "v"(lds_addr), "v"(gaddr) : "memory");
}

#if __has_builtin(__builtin_amdgcn_s_wait_asynccnt)
#define WAIT_ASYNC0() __builtin_amdgcn_s_wait_asynccnt(0)
#else
#define WAIT_ASYNC0() asm volatile("s_wait_asynccnt 0x0" ::: "memory")
#endif

// ---------------------------------------------------------------------------
// Kernel 1: flash-decode partial attention. One wave32 per workgroup.
// grid = (B*H, NSPLIT), block = 32.
// Split 7 covers only keys [3584, 4080) in the tile loop; the appended key
// (index 4095) is folded in by a scalar epilogue from LDS.
// ---------------------------------------------------------------------------
__global__ __launch_bounds__(32) void attn_partial_kernel(
    const float* __restrict__ q, const float* __restrict__ k,
    const float* __restrict__ v, const float* __restrict__ cache_k,
    const float* __restrict__ cache_v,
    float* __restrict__ part_acc,   // [512*NSPLIT][128]
    float* __restrict__ part_ml)    // [512*NSPLIT][2]  (m, l)
{
    __shared__ __align__(16) float sQ[D_];                 // roped, pre-scaled q
    __shared__ __align__(16) float sKn[D_];                // roped new k
    __shared__ __align__(16) float sK[2][16 * SK_PITCH];   // double-buffered tile

    const int lane  = threadIdx.x;       // 0..31
    const int bh    = blockIdx.x;        // 0..511
    const int split = blockIdx.y;        // 0..NSPLIT-1
    const int b     = bh >> 5;
    const int h     = bh & (H_ - 1);

    const float* qp = q + (size_t)(b * H_ + h) * D_;
    const float* kp = k + (size_t)(b * H_ + h) * D_;
    const float* vp = v + (size_t)(b * H_ + h) * D_;
    const float* Kc = cache_k + (size_t)b * PAST_ * ROWSTRIDE + (size_t)h * D_;
    const float* Vc = cache_v + (size_t)b * PAST_ * ROWSTRIDE + (size_t)h * D_;

    const bool last_split = (split == NSPLIT - 1);
    const int  ntiles     = last_split ? TILES_PER_SPLIT - 1 : TILES_PER_SPLIT;

    const unsigned sKbase = (unsigned)(size_t)&sK[0][0];   // LDS byte offset

    // Async-prefetch one 16x128 K tile (all rows cache-resident by design).
    auto stage_tile = [&](int tile, int bufsel) {
        const int      key_base = split * KEYS_PER_SPLIT + tile * 16;
        const unsigned lbase    = sKbase + (unsigned)bufsel * SK_BUFBYTES
                                         + (unsigned)lane * 16u;
        const float*   g        = Kc + (size_t)key_base * ROWSTRIDE + lane * 4;
#pragma unroll
        for (int r = 0; r < 16; ++r)
            async_ld_b128(lbase + (unsigned)(r * SK_PITCH * 4),
                          g + (size_t)r * ROWSTRIDE);
    };

    // Kick off tile 0 before doing RoPE, so HBM fetch overlaps the math.
    stage_tile(0, 0);

    // ---- RoPE: lane handles j = 2*lane, 2*lane+1  (j in [0,64)) ----
    const float scale = 0.08838834764831845f;  // 1/sqrt(128), folded into q
    const float pos   = (float)PAST_;          // position of the new token
#pragma unroll
    for (int t = 0; t < 2; ++t) {
        int   j        = 2 * lane + t;
        float inv_freq = __expf(-0.14391156515f * (float)j); // 10000^(-j/64)
        float ang      = pos * inv_freq;
        float cs = __cosf(ang), sn = __sinf(ang);
        float q1 = qp[j], q2 = qp[j + 64];
        sQ[j]      = (q1 * cs - q2 * sn) * scale;
        sQ[j + 64] = (q2 * cs + q1 * sn) * scale;
        float k1 = kp[j], k2 = kp[j + 64];
        sKn[j]      = k1 * cs - k2 * sn;
        sKn[j + 64] = k2 * cs + k1 * sn;
    }
    __syncthreads();

    // WMMA fragment slot geometry (A 16x4 / B 4x16, f32):
    //   lanes 0-15  hold K-dims {0,1}, lanes 16-31 hold K-dims {2,3}
    const int hi  = (lane >= 16) ? 2 : 0;
    const int n16 = lane & 15;             // B-matrix column / key-in-tile

    float m_run = -__builtin_inff();
    float l_run = 0.0f;
    float acc0 = 0.f, acc1 = 0.f, acc2 = 0.f, acc3 = 0.f;  // dims 4*lane..+3

    for (int t = 0; t < ntiles; ++t) {
        const int key_base = split * KEYS_PER_SPLIT + t * 16;

        WAIT_ASYNC0();                       // tile t resident in sK[t&1]
        if (t + 1 < ntiles)
            stage_tile(t + 1, (t + 1) & 1);  // prefetch next tile

        // ---- QK^T: 32 x V_WMMA_F32_16X16X4_F32, K=128 accumulated ----
        // A[m][k] = qhat[c*4+k] (broadcast over m); B[k][n] = K[n][c*4+k]
        // => D[m][n] = score[n] for all m; lane l holds score[l%16].
        v8f cfrag = {};
        const float* bk = &sK[t & 1][n16 * SK_PITCH];
#pragma unroll
        for (int c = 0; c < 32; ++c) {
            v2f a  = *(const v2f*)(sQ + c * 4 + hi);
            v2f bb = *(const v2f*)(bk + c * 4 + hi);
            cfrag = __builtin_amdgcn_wmma_f32_16x16x4_f32(
                false, a, false, bb, (short)0, cfrag, false, false);
        }
        float sc = cfrag[0];   // all 8 rows identical; pre-scaled by 1/sqrt(D)

        // ---- online softmax over this 16-key tile ----
        float tmax = sc;
#pragma unroll
        for (int off = 1; off < 16; off <<= 1)
            tmax = fmaxf(tmax, __shfl_xor(tmax, off, 32));
        float m_new = fmaxf(m_run, tmax);
        float p     = __expf(sc - m_new);
        float fac   = __expf(m_run - m_new);   // exp(-inf)=0 on first tile
        float psum  = p;
#pragma unroll
        for (int off = 1; off < 16; off <<= 1)
            psum += __shfl_xor(psum, off, 32); // lanes 0-15 and 16-31 mirror
        l_run = l_run * fac + psum;
        acc0 *= fac; acc1 *= fac; acc2 *= fac; acc3 *= fac;

        // ---- P*V: lane owns dims [4*lane, 4*lane+3], streaming b128 ----
        const float* vtile = Vc + (size_t)key_base * ROWSTRIDE + lane * 4;
#pragma unroll
        for (int n = 0; n < 16; ++n) {
            float pn = __shfl(p, n, 32);
            v4f vv = __builtin_nontemporal_load(
                         (const v4f*)(vtile + (size_t)n * ROWSTRIDE));
            acc0 += pn * vv.x; acc1 += pn * vv.y;
            acc2 += pn * vv.z; acc3 += pn * vv.w;
        }
        m_run = m_new;
    }

    // ---- epilogue (split 7 only): fold in the appended key (index 4095) ----
    if (last_split) {
        v4f qv = *(const v4f*)(&sQ[lane * 4]);
        v4f kv = *(const v4f*)(&sKn[lane * 4]);
        float d = qv.x * kv.x + qv.y * kv.y + qv.z * kv.z + qv.w * kv.w;
#pragma unroll
        for (int off = 1; off < 32; off <<= 1)
            d += __shfl_xor(d, off, 32);       // full-wave dot(sQ, sKn)
        float m_new = fmaxf(m_run, d);
        float fac   = __expf(m_run - m_new);
        float pn    = __expf(d - m_new);
        l_run = l_run * fac + pn;
        v4f vv = *(const v4f*)(vp + lane * 4);  // appended v (no RoPE)
        acc0 = acc0 * fac + pn * vv.x;
        acc1 = acc1 * fac + pn * vv.y;
        acc2 = acc2 * fac + pn * vv.z;
        acc3 = acc3 * fac + pn * vv.w;
        m_run = m_new;
    }

    // ---- emit partial (m, l, acc[128]) ----
    const int part = bh * NSPLIT + split;
    v4f out4 = { acc0, acc1, acc2, acc3 };
    *(v4f*)(part_acc + (size_t)part * D_ + lane * 4) = out4;
    if (lane == 0) {
        part_ml[2 * part + 0] = m_run;
        part_ml[2 * part + 1] = l_run;
    }
}

// ---------------------------------------------------------------------------
// Kernel 2: merge the NSPLIT partials per (b,h). grid = 512, block = 128.
// ---------------------------------------------------------------------------
__global__ __launch_bounds__(128) void attn_combine_kernel(
    const float* __restrict__ part_acc, const float* __restrict__ part_ml,
    float* __restrict__ out)
{
    const int bh = blockIdx.x;
    const int d  = threadIdx.x;

    float ms[NSPLIT], ls[NSPLIT];
    float M = -__builtin_inff();
#pragma unroll
    for (int i = 0; i < NSPLIT; ++i) {
        ms[i] = part_ml[2 * (bh * NSPLIT + i) + 0];
        ls[i] = part_ml[2 * (bh * NSPLIT + i) + 1];
        M = fmaxf(M, ms[i]);
    }
    float S = 0.f, o = 0.f;
#pragma unroll
    for (int i = 0; i < NSPLIT; ++i) {
        float w = __expf(ms[i] - M);
        S += ls[i] * w;
        o += part_acc[(size_t)(bh * NSPLIT + i) * D_ + d] * w;
    }
    out[(size_t)bh * D_ + d] = o / S;   // out[b, h*128 + d]
}

// ---------------------------------------------------------------------------
extern "C" void kernel_launch(void* const* d_in, const int* in_sizes, int n_in,
                              void* d_out, int out_size, void* d_ws, size_t ws_size,
                              hipStream_t stream) {
    const float* q       = (const float*)d_in[0];
    const float* k       = (const float*)d_in[1];
    const float* v       = (const float*)d_in[2];
    const float* cache_k = (const float*)d_in[3];
    const float* cache_v = (const float*)d_in[4];
    // d_in[5] = past_len (compile-time constant 4095 here)

    float* part_acc = (float*)d_ws;                                  // 512*8*128 f
    float* part_ml  = part_acc + (size_t)(B_ * H_) * NSPLIT * D_;    // 512*8*2 f
    float* out      = (float*)d_out;

    dim3 grid1(B_ * H_, NSPLIT);
    attn_partial_kernel<<<grid1, 32, 0, stream>>>(q, k, v, cache_k, cache_v,
                                                  part_acc, part_ml);
    attn_combine_kernel<<<B_ * H_, D_, 0, stream>>>(part_acc, part_ml, out);
}